// GCN_39977555591298
// MI455X (gfx1250) — compile-verified
//
#include <hip/hip_runtime.h>
#include <hip/hip_bf16.h>

// ---------------------------------------------------------------------------
// GCN inference pipeline for MI455X (gfx1250, wave32).
// Dense GEMMs use V_WMMA_F32_16X16X4_F32 (fp32 matrix op -> matches reference
// precision; pipeline is L2/atomic bound so low-precision WMMA buys nothing).
// Weights are pre-transposed into workspace so both A and B fragments are
// contiguous global_load_b64 per K-step.
// ---------------------------------------------------------------------------

typedef float v2f __attribute__((ext_vector_type(2)));
typedef float v8f __attribute__((ext_vector_type(8)));

#define GCN_H 128

__global__ void k_fill(float* __restrict__ p, float v, int n) {
    int i = blockIdx.x * blockDim.x + threadIdx.x;
    if (i < n) p[i] = v;
}

__global__ void k_deg_count(const long long* __restrict__ dst,
                            float* __restrict__ deg, int e) {
    int i = blockIdx.x * blockDim.x + threadIdx.x;
    if (i < e) {
        __hip_atomic_fetch_add(&deg[(int)dst[i]], 1.0f,
                               __ATOMIC_RELAXED, __HIP_MEMORY_SCOPE_AGENT);
    }
}

__global__ void k_norm(const float* __restrict__ deg,
                       float* __restrict__ dis,
                       float* __restrict__ selfc, int n) {
    int i = blockIdx.x * blockDim.x + threadIdx.x;
    if (i < n) {
        float d = deg[i];
        dis[i]   = rsqrtf(d);
        selfc[i] = 1.0f / d;     // dis*dis
    }
}

__global__ void k_coef(const long long* __restrict__ src,
                       const long long* __restrict__ dst,
                       const float* __restrict__ dis,
                       float* __restrict__ coef, int e) {
    int i = blockIdx.x * blockDim.x + threadIdx.x;
    if (i < e) coef[i] = dis[(int)src[i]] * dis[(int)dst[i]];
}

// Wt[c][k] = W[k][c]   (rows = k-dim of W, cols = n-dim of W)
__global__ void k_transpose(const float* __restrict__ W,
                            float* __restrict__ Wt, int rows, int cols) {
    int i = blockIdx.x * blockDim.x + threadIdx.x;
    if (i < rows * cols) {
        const int r = i / cols;
        const int c = i - r * cols;
        Wt[(size_t)c * rows + r] = W[i];
    }
}

// ---------------------------------------------------------------------------
// WMMA GEMM: D = A[rows x kdim] * W[kdim x ncols], W given TRANSPOSED
// (Wt[ncols x kdim]) so the B fragment is a contiguous b64 load per K-step.
// mode 0: t = D ; agg = D * selfc[row] + bias[col]   (GCN layer pre-scatter)
// mode 1: out = D + bias[col]                         (final classifier)
// Grid: rows/16 blocks; blockDim.x = (ncols/16)*32 -> one wave per col tile.
// No divergence before WMMA -> EXEC all ones as required by the ISA.
// ---------------------------------------------------------------------------
__global__ void k_wmma_gemm(const float* __restrict__ A,
                            const float* __restrict__ Wt,
                            const float* __restrict__ bias,
                            const float* __restrict__ selfc,
                            float* __restrict__ t,
                            float* __restrict__ agg,
                            int kdim, int ncols, int mode) {
    const int lane = threadIdx.x & 31;
    const int wave = threadIdx.x >> 5;
    const int row0 = blockIdx.x << 4;
    const int col0 = wave << 4;
    const int half = lane >> 4;   // 0: lanes 0-15, 1: lanes 16-31
    const int l16  = lane & 15;

    // A 16x4 fragment: lane holds row M=l16, K = 2*half + {0,1}
    const float* arow = A + (size_t)(row0 + l16) * kdim + 2 * half;
    // B 4x16 fragment: lane holds col N=l16, K = 2*half + {0,1}
    // Wt is [ncols][kdim] -> contiguous pair per lane.
    const float* brow = Wt + (size_t)(col0 + l16) * kdim + 2 * half;

    v8f acc = {0.f, 0.f, 0.f, 0.f, 0.f, 0.f, 0.f, 0.f};
    for (int k = 0; k < kdim; k += 4) {
        const v2f a = *(const v2f*)(arow + k);
        const v2f b = *(const v2f*)(brow + k);
        acc = __builtin_amdgcn_wmma_f32_16x16x4_f32(
            /*neg_a=*/false, a, /*neg_b=*/false, b,
            /*c_mod=*/(short)0, acc, /*reuse_a=*/false, /*reuse_b=*/false);
    }

    // C/D layout: lanes 0-15 -> M = v (VGPR idx); lanes 16-31 -> M = 8 + v
    const int col = col0 + l16;
    const float bv = bias[col];
    if (mode == 0) {
#pragma unroll
        for (int v = 0; v < 8; ++v) {
            const int row = row0 + (half << 3) + v;
            const float val = acc[v];
            t[(size_t)row * ncols + col]   = val;
            agg[(size_t)row * ncols + col] = fmaf(val, selfc[row], bv);
        }
    } else {
#pragma unroll
        for (int v = 0; v < 8; ++v) {
            const int row = row0 + (half << 3) + v;
            agg[(size_t)row * ncols + col] = acc[v] + bv;
        }
    }
}

// ---------------------------------------------------------------------------
// Edge scatter: agg[dst] += t[src] * coef.  One thread per (edge, 4 features):
// consecutive threads cover one edge's 512B row (coalesced float4 gather),
// then 4 relaxed agent-scope fp32 atomic adds (resolve in the 192MB L2 where
// t and agg are resident).
// ---------------------------------------------------------------------------
__global__ void k_scatter(const long long* __restrict__ src,
                          const long long* __restrict__ dst,
                          const float* __restrict__ coef,
                          const float* __restrict__ t,
                          float* __restrict__ agg, int e) {
    int tid = blockIdx.x * blockDim.x + threadIdx.x;
    if (tid >= e * 32) return;           // 128 feats / 4 per thread = 32
    const int ei = tid >> 5;
    const int f  = (tid & 31) << 2;
    const int s  = (int)src[ei];
    const int d  = (int)dst[ei];
    const float cf = coef[ei];
    const float4 v = *(const float4*)(t + (size_t)s * GCN_H + f);
    float* ap = agg + (size_t)d * GCN_H + f;
    __hip_atomic_fetch_add(ap + 0, v.x * cf, __ATOMIC_RELAXED, __HIP_MEMORY_SCOPE_AGENT);
    __hip_atomic_fetch_add(ap + 1, v.y * cf, __ATOMIC_RELAXED, __HIP_MEMORY_SCOPE_AGENT);
    __hip_atomic_fetch_add(ap + 2, v.z * cf, __ATOMIC_RELAXED, __HIP_MEMORY_SCOPE_AGENT);
    __hip_atomic_fetch_add(ap + 3, v.w * cf, __ATOMIC_RELAXED, __HIP_MEMORY_SCOPE_AGENT);
}

__global__ void k_bn_relu(const float* __restrict__ agg,
                          const float* __restrict__ gamma,
                          const float* __restrict__ beta,
                          const float* __restrict__ mean,
                          const float* __restrict__ var,
                          float* __restrict__ out, int total) {
    int i = blockIdx.x * blockDim.x + threadIdx.x;
    if (i >= total) return;
    const int c = i & (GCN_H - 1);
    const float inv = rsqrtf(var[c] + 1e-5f);
    const float y = (agg[i] - mean[c]) * inv * gamma[c] + beta[c];
    out[i] = fmaxf(y, 0.0f);
}

extern "C" void kernel_launch(void* const* d_in, const int* in_sizes, int n_in,
                              void* d_out, int out_size, void* d_ws, size_t ws_size,
                              hipStream_t stream) {
    (void)n_in; (void)out_size; (void)ws_size;
    const int IN = 128, H = 128, C = 64, L = 5;
    const int n = in_sizes[0] / IN;      // 100000 (multiple of 16)
    const int e = in_sizes[1] / 2;       // 640000

    const float*     x    = (const float*)d_in[0];
    const long long* ei   = (const long long*)d_in[1];   // int64 (2,E)
    const long long* esrc = ei;
    const long long* edst = ei + e;
    const float* W_in  = (const float*)d_in[2];
    const float* b_in  = (const float*)d_in[3];
    const float* Ws    = (const float*)d_in[4];          // (L-1,H,H)
    const float* bs    = (const float*)d_in[5];          // (L-1,H)
    const float* gam   = (const float*)d_in[6];
    const float* bet   = (const float*)d_in[7];
    const float* rmean = (const float*)d_in[8];
    const float* rvar  = (const float*)d_in[9];
    const float* W_out = (const float*)d_in[10];
    const float* b_out = (const float*)d_in[11];

    // workspace carve-up (floats)
    float* ws    = (float*)d_ws;
    float* deg   = ws;
    float* dis   = deg + n;
    float* selfc = dis + n;
    float* coef  = selfc + n;
    float* t     = coef + e;
    float* agg   = t + (size_t)n * H;
    float* hA    = agg + (size_t)n * H;
    float* hB    = hA + (size_t)n * H;
    float* wT    = hB + (size_t)n * H;   // 5*H*H + H*C floats of transposed W

    const int T = 256;
    // ---- transpose all weight matrices once (tiny vs pipeline traffic) ----
    for (int l = 0; l < L; ++l) {
        const float* W = (l == 0) ? W_in : Ws + (size_t)(l - 1) * H * H;
        k_transpose<<<(H * H + T - 1) / T, T, 0, stream>>>(W, wT + (size_t)l * H * H, H, H);
    }
    float* WoT = wT + (size_t)L * H * H;
    k_transpose<<<(H * C + T - 1) / T, T, 0, stream>>>(W_out, WoT, H, C);

    // ---- normalization coefficients ----
    k_fill<<<(n + T - 1) / T, T, 0, stream>>>(deg, 1.0f, n);
    k_deg_count<<<(e + T - 1) / T, T, 0, stream>>>(edst, deg, e);
    k_norm<<<(n + T - 1) / T, T, 0, stream>>>(deg, dis, selfc, n);
    k_coef<<<(e + T - 1) / T, T, 0, stream>>>(esrc, edst, dis, coef, e);

    // ---- 5 GCN layers ----
    const float* cur = x;
    float* nxt = hA;
    const int nh = n * H;
    for (int l = 0; l < L; ++l) {
        const float* b = (l == 0) ? b_in : bs + (size_t)(l - 1) * H;
        k_wmma_gemm<<<n / 16, (H / 16) * 32, 0, stream>>>(
            cur, wT + (size_t)l * H * H, b, selfc, t, agg, H, H, 0);
        k_scatter<<<(e * 32 + T - 1) / T, T, 0, stream>>>(esrc, edst, coef, t, agg, e);
        k_bn_relu<<<(nh + T - 1) / T, T, 0, stream>>>(
            agg, gam + (size_t)l * H, bet + (size_t)l * H,
            rmean + (size_t)l * H, rvar + (size_t)l * H, nxt, nh);
        cur = nxt;
        nxt = (nxt == hA) ? hB : hA;
    }

    // ---- final classifier: [N,H] @ [H,C] + b ----
    k_wmma_gemm<<<n / 16, (C / 16) * 32, 0, stream>>>(
        cur, WoT, b_out, /*selfc*/nullptr, /*t*/nullptr,
        (float*)d_out, H, C, 1);
}